// DTCWTInverse_44126493999649
// MI455X (gfx1250) — compile-verified
//
#include <hip/hip_runtime.h>
#include <cstddef>
#include <cstdint>

// DTCWT 1-D inverse for MI455X (gfx1250, wave32).
// Memory-bound (~0.75 GB traffic -> ~33us at 23.3 TB/s); compute done in fp32
// on the matrix pipe via V_WMMA_F32_16X16X4_F32 (preserves fp32 semantics,
// unlike f16/bf16 WMMA shapes).

#define B_ 32
#define C_ 4
#define T_ 262144
#define J_ 8
#define SQRT2F 1.41421356237309515f

typedef __attribute__((ext_vector_type(2))) float v2f;
typedef __attribute__((ext_vector_type(8))) float v8f;

#if __has_builtin(__builtin_amdgcn_wmma_f32_16x16x4_f32)
#define USE_WMMA_F32 1
#else
#define USE_WMMA_F32 0
#endif

// ---------------------------------------------------------------------------
// One coarse synthesis level:
//   out = colifilt(lo, g0b, g0a) + colifilt(hi, g1b, g1a),
//   hi[2m] = sqrt2*r[m], hi[2m+1] = sqrt2*i[m].
// Derivation from the reference (t = per-channel length of lo == length of hi,
// th = t/2 = length of r/i; output length 2t):
//   out[4s+0] = sum_k hL [2k]  *x[2s+4-2k] + hH [2k]  *hi[2s+4-2k]
//   out[4s+2] = sum_k hL [2k+1]*x[2s+4-2k] + hH [2k+1]*hi[2s+4-2k]
//   out[4s+1] = sum_k hLb[2k]  *x[2s+5-2k] + hHb[2k]  *hi[2s+5-2k]
//   out[4s+3] = sum_k hLb[2k+1]*x[2s+5-2k] + hHb[2k+1]*hi[2s+5-2k]
// with hL=g0b, hLb=g0a, hH=g1b, hHb=g1a, k=0..4, zero padding out of range.
//
// WMMA tile: 16 consecutive groups s0..s0+15 (rows M), phases in columns 0..3.
// Five chained V_WMMA_F32_16X16X4_F32 accumulate taps 0..3 of each of the four
// data windows plus a tap-4 cleanup matmul.
// ---------------------------------------------------------------------------
__global__ __launch_bounds__(256) void dtcwt_colifilt_wmma(
    const float* __restrict__ lo, const float* __restrict__ bpr,
    const float* __restrict__ bpi,
    const float* __restrict__ g0b, const float* __restrict__ g0a,
    const float* __restrict__ g1b, const float* __restrict__ g1a,
    float* __restrict__ out, int t)
{
  const int wave = blockIdx.x * (blockDim.x >> 5) + (threadIdx.x >> 5);
  const int lane = threadIdx.x & 31;
  const int tilesPerBC = t >> 5;              // (t/2)/16 output groups per tile
  const int bc = wave / tilesPerBC;
  const int s0 = (wave - bc * tilesPerBC) << 4;
  const int th = t >> 1;

  const float* __restrict__ x  = lo  + (size_t)bc * t;
  const float* __restrict__ rr = bpr + (size_t)bc * th;
  const float* __restrict__ ii = bpi + (size_t)bc * th;

  // Filter taps: uniform addresses -> scalar loads.
  float hL[10], hLb[10], hH[10], hHb[10];
#pragma unroll
  for (int k = 0; k < 10; ++k) {
    hL[k] = g0b[k]; hLb[k] = g0a[k]; hH[k] = g1b[k]; hHb[k] = g1a[k];
  }

  const int m    = lane & 15;
  const int half = lane >> 4;

  auto LDX = [&](int idx) -> float {
    return ((unsigned)idx < (unsigned)t)  ? x[idx]            : 0.f; };
  auto LDR = [&](int idx) -> float {
    return ((unsigned)idx < (unsigned)th) ? SQRT2F * rr[idx]  : 0.f; };
  auto LDI = [&](int idx) -> float {
    return ((unsigned)idx < (unsigned)th) ? SQRT2F * ii[idx]  : 0.f; };

#if USE_WMMA_F32
  // A layout (16x4 f32, ISA 7.12.2): lanes 0-15 hold K=0,1; lanes 16-31 K=2,3.
  const int s  = s0 + m;
  const int k0 = half << 1, k1 = k0 + 1;

  v2f A1; A1.x = LDX(2*s + 4 - 2*k0); A1.y = LDX(2*s + 4 - 2*k1); // x even win
  v2f A2; A2.x = LDR(s + 2 - k0);     A2.y = LDR(s + 2 - k1);     // hi even win
  v2f A3; A3.x = LDX(2*s + 5 - 2*k0); A3.y = LDX(2*s + 5 - 2*k1); // x odd win
  v2f A4; A4.x = LDI(s + 2 - k0);     A4.y = LDI(s + 2 - k1);     // hi odd win
  v2f A5;                                                          // tap 4
  if (half == 0) { A5.x = LDX(2*s - 4); A5.y = LDR(s - 2); }       // rows 0,1
  else           { A5.x = LDX(2*s - 3); A5.y = LDI(s - 2); }       // rows 2,3

  // B layout (4x16 f32): VGPR0 -> rows {0,2} across lane halves, VGPR1 -> {1,3}.
  const int n = m;
  v2f B1; B1.x = (n==0)?hL[4*half]    : (n==2)?hL[4*half+1]   : 0.f;
          B1.y = (n==0)?hL[4*half+2]  : (n==2)?hL[4*half+3]   : 0.f;
  v2f B2; B2.x = (n==0)?hH[4*half]    : (n==2)?hH[4*half+1]   : 0.f;
          B2.y = (n==0)?hH[4*half+2]  : (n==2)?hH[4*half+3]   : 0.f;
  v2f B3; B3.x = (n==1)?hLb[4*half]   : (n==3)?hLb[4*half+1]  : 0.f;
          B3.y = (n==1)?hLb[4*half+2] : (n==3)?hLb[4*half+3]  : 0.f;
  v2f B4; B4.x = (n==1)?hHb[4*half]   : (n==3)?hHb[4*half+1]  : 0.f;
          B4.y = (n==1)?hHb[4*half+2] : (n==3)?hHb[4*half+3]  : 0.f;
  v2f B5;
  B5.x = (half==0) ? ((n==0)?hL[8] :(n==2)?hL[9] :0.f)
                   : ((n==1)?hLb[8]:(n==3)?hLb[9]:0.f);
  B5.y = (half==0) ? ((n==0)?hH[8] :(n==2)?hH[9] :0.f)
                   : ((n==1)?hHb[8]:(n==3)?hHb[9]:0.f);

  v8f c = {0.f,0.f,0.f,0.f,0.f,0.f,0.f,0.f};
  c = __builtin_amdgcn_wmma_f32_16x16x4_f32(false, A1, false, B1, (short)0, c, false, false);
  c = __builtin_amdgcn_wmma_f32_16x16x4_f32(false, A3, false, B3, (short)0, c, false, false);
  c = __builtin_amdgcn_wmma_f32_16x16x4_f32(false, A2, false, B2, (short)0, c, false, false);
  c = __builtin_amdgcn_wmma_f32_16x16x4_f32(false, A4, false, B4, (short)0, c, false, false);
  c = __builtin_amdgcn_wmma_f32_16x16x4_f32(false, A5, false, B5, (short)0, c, false, false);

  // D layout: c[v] holds D[v + 8*half][n]; only columns n<4 are real outputs.
  if (n < 4) {
    float* o = out + (size_t)bc * (size_t)(2*t) + (size_t)(4*s0 + 32*half + n);
#pragma unroll
    for (int v = 0; v < 8; ++v) o[4*v] = c[v];
  }
#else
  // Fallback: direct fp32 FMA (lanes 0-15 each own one output group).
  if (half == 0) {
    const int s = s0 + m;
    float p0 = 0.f, p1 = 0.f, p2 = 0.f, p3 = 0.f;
#pragma unroll
    for (int k = 0; k < 5; ++k) {
      float xe = LDX(2*s + 4 - 2*k), xo = LDX(2*s + 5 - 2*k);
      float he = LDR(s + 2 - k),     ho = LDI(s + 2 - k);
      p0 = fmaf(hL[2*k],    xe, fmaf(hH[2*k],    he, p0));
      p2 = fmaf(hL[2*k+1],  xe, fmaf(hH[2*k+1],  he, p2));
      p1 = fmaf(hLb[2*k],   xo, fmaf(hHb[2*k],   ho, p1));
      p3 = fmaf(hLb[2*k+1], xo, fmaf(hHb[2*k+1], ho, p3));
    }
    float* o = out + (size_t)bc * (size_t)(2*t) + 4*(size_t)s;
    o[0] = p0; o[1] = p1; o[2] = p2; o[3] = p3;
  }
#endif
}

// ---------------------------------------------------------------------------
// Finest level: out[n] = sum_{k=0..6} g0o[k]*lo[n+3-k]
//                      + sum_{k=0..4} g1o[k]*hi[n+2-k],
// hi[j] = sqrt2 * (j even ? r[j/2] : i[j/2]), zero padding. Pure VALU; the
// conv is 12 FMAs/output and entirely HBM-bandwidth limited.
// ---------------------------------------------------------------------------
__global__ __launch_bounds__(256) void dtcwt_colfilter_j1(
    const float* __restrict__ lo, const float* __restrict__ bpr,
    const float* __restrict__ bpi,
    const float* __restrict__ g0o, const float* __restrict__ g1o,
    float* __restrict__ out)
{
  const size_t gid = (size_t)blockIdx.x * blockDim.x + threadIdx.x;
  const int bc   = (int)(gid >> 18);            // / T_
  const int nidx = (int)(gid & (size_t)(T_ - 1));
  const float* __restrict__ L = lo  + ((size_t)bc << 18);
  const float* __restrict__ R = bpr + ((size_t)bc << 17);
  const float* __restrict__ I = bpi + ((size_t)bc << 17);

  float h0[7], h1[5];
#pragma unroll
  for (int k = 0; k < 7; ++k) h0[k] = g0o[k];
#pragma unroll
  for (int k = 0; k < 5; ++k) h1[k] = g1o[k];

  float acc = 0.f;
#pragma unroll
  for (int k = 0; k < 7; ++k) {
    int j = nidx + 3 - k;
    if ((unsigned)j < (unsigned)T_) acc = fmaf(h0[k], L[j], acc);
  }
#pragma unroll
  for (int k = 0; k < 5; ++k) {
    int j = nidx + 2 - k;
    if ((unsigned)j < (unsigned)T_) {
      float hv = (j & 1) ? I[j >> 1] : R[j >> 1];
      acc = fmaf(h1[k], SQRT2F * hv, acc);
    }
  }
  out[gid] = acc;
}

// ---------------------------------------------------------------------------
// Host-side orchestration: 7 colifilt levels ping-ponging through d_ws
// (region0: B*C*T floats = 128MB for the j=1 output; region1: <=64MB for the
// even-level outputs), then the final colfilter level into d_out.
// Input order (setup_inputs dict): 0=yl, 1+2j / 2+2j = yh{j}_r / yh{j}_i,
// 17=g0o, 18=g1o, 19=g0a, 20=g0b, 21=g1a, 22=g1b.
// ---------------------------------------------------------------------------
extern "C" void kernel_launch(void* const* d_in, const int* in_sizes, int n_in,
                              void* d_out, int out_size, void* d_ws, size_t ws_size,
                              hipStream_t stream)
{
  (void)in_sizes; (void)n_in; (void)out_size; (void)ws_size;

  const float* yl  = (const float*)d_in[0];
  const float* g0o = (const float*)d_in[17];
  const float* g1o = (const float*)d_in[18];
  const float* g0a = (const float*)d_in[19];
  const float* g0b = (const float*)d_in[20];
  const float* g1a = (const float*)d_in[21];
  const float* g1b = (const float*)d_in[22];

  float* ws0 = (float*)d_ws;
  float* ws1 = ws0 + (size_t)B_ * C_ * T_;   // +128MB

  const float* cur = yl;
  int t = T_ >> (J_ - 1);                    // 2048
  int flip = 0;
  for (int j = J_ - 1; j >= 1; --j) {
    const float* r = (const float*)d_in[1 + 2*j];
    const float* i = (const float*)d_in[2 + 2*j];
    float* nxt = flip ? ws1 : ws0;
    // tiles = B*C*(t/2)/16 = 4t; 8 waves per 256-thread block -> t/2 blocks.
    dim3 grid((unsigned)(t >> 1)), block(256);
    dtcwt_colifilt_wmma<<<grid, block, 0, stream>>>(
        cur, r, i, g0b, g0a, g1b, g1a, nxt, t);
    cur = nxt;
    flip ^= 1;
    t <<= 1;
  }
  // cur == ws0 (length B*C*T), t == T_.
  {
    dim3 grid((unsigned)(((size_t)B_ * C_ * T_) / 256)), block(256);
    dtcwt_colfilter_j1<<<grid, block, 0, stream>>>(
        cur, (const float*)d_in[1], (const float*)d_in[2], g0o, g1o,
        (float*)d_out);
  }
}